// GATEncoder_12309376270478
// MI455X (gfx1250) — compile-verified
//
#include <hip/hip_runtime.h>

typedef float v2f __attribute__((ext_vector_type(2)));
typedef float v8f __attribute__((ext_vector_type(8)));

constexpr int N_NODES = 100000;
constexpr int N_EDGES = 1600000;
constexpr int IN_C = 128;
constexpr int HID  = 64;
constexpr int LAT  = 32;
constexpr int L    = 3;
constexpr int MTILES = N_NODES / 16;          // 6250 (exact)
constexpr int WAVES_PER_BLOCK = 8;            // 256 threads = 8 wave32
constexpr int GEMM_GX = (MTILES + WAVES_PER_BLOCK - 1) / WAVES_PER_BLOCK;

// ---------------------------------------------------------------------------
// WMMA fp32 GEMM:  C[N, COLS] = A[N, K] @ W[K, COLS] (+ bias) (+= if ACC)
// One wave computes one 16x16 output tile via V_WMMA_F32_16X16X4_F32.
//  A 16x4 layout: lane&15 = M row; lane>>4 selects K pair; v[0..1] = K,K+1
//  B 4x16 layout (mirror): lane&15 = N col; lane>>4 selects K pair
//  C/D 16x16:  VGPR r -> M = r + 8*(lane>>4), N = lane&15
// ---------------------------------------------------------------------------
template <int K, int COLS, bool ACC>
__global__ __launch_bounds__(256)
void wmma_gemm_kernel(const float* __restrict__ A, const float* __restrict__ W,
                      const float* __restrict__ bias, float* __restrict__ C)
{
    const int lane  = threadIdx.x & 31;
    const int wave  = threadIdx.x >> 5;
    const int mtile = blockIdx.x * WAVES_PER_BLOCK + wave;
    if (mtile >= MTILES) return;

    const int m0   = mtile * 16;
    const int n0   = blockIdx.y * 16;
    const int half = lane >> 4;     // 0 or 1
    const int l15  = lane & 15;

    v8f acc = {0.f, 0.f, 0.f, 0.f, 0.f, 0.f, 0.f, 0.f};

    const float* __restrict__ arow = A + (size_t)(m0 + l15) * K + 2 * half;
    const float* __restrict__ bcol = W + (size_t)(2 * half) * COLS + (n0 + l15);

#pragma unroll 8
    for (int k0 = 0; k0 < K; k0 += 4) {
        v2f a, b;
        a[0] = arow[0];
        a[1] = arow[1];
        b[0] = bcol[0];
        b[1] = bcol[COLS];
        acc = __builtin_amdgcn_wmma_f32_16x16x4_f32(
            /*neg_a=*/false, a, /*neg_b=*/false, b,
            /*c_mod=*/(short)0, acc, /*reuse_a=*/false, /*reuse_b=*/false);
        arow += 4;
        bcol += 4 * COLS;
    }

    const int col = n0 + l15;
    const float bv = bias ? bias[col] : 0.0f;
#pragma unroll
    for (int r = 0; r < 8; ++r) {
        const int row = m0 + r + half * 8;
        float v = acc[r] + bv;
        if (ACC) v += C[(size_t)row * COLS + col];
        C[(size_t)row * COLS + col] = v;
    }
}

// ---------------------------------------------------------------------------
// Degree / normalization kernels
// ---------------------------------------------------------------------------
__global__ void deg_init_kernel(float* __restrict__ deg)
{
    int i = blockIdx.x * blockDim.x + threadIdx.x;
    if (i < N_NODES) deg[i] = 1.0f;   // self loop
}

__global__ void deg_count_kernel(float* __restrict__ deg, const int* __restrict__ dst)
{
    int e = blockIdx.x * blockDim.x + threadIdx.x;
    if (e < N_EDGES) atomicAdd(&deg[dst[e]], 1.0f);
}

__global__ void deg_rsqrt_kernel(float* __restrict__ dinv)
{
    int i = blockIdx.x * blockDim.x + threadIdx.x;
    if (i < N_NODES) dinv[i] = rsqrtf(dinv[i]);   // deg >= 1 always (self loop)
}

__global__ void edge_norm_kernel(float* __restrict__ norm, const float* __restrict__ dinv,
                                 const int* __restrict__ src, const int* __restrict__ dst)
{
    int e = blockIdx.x * blockDim.x + threadIdx.x;
    if (e < N_EDGES) norm[e] = dinv[src[e]] * dinv[dst[e]];
}

// ---------------------------------------------------------------------------
// Aggregation: out = segment_sum(hw[src]*norm, dst) + hw*dinv^2 (self loop) + b
// Split: init pass (self loop + bias, full overwrite) then atomic edge pass.
// ---------------------------------------------------------------------------
template <int C>
__global__ void agg_init_kernel(float* __restrict__ out, const float* __restrict__ hw,
                                const float* __restrict__ dinv, const float* __restrict__ bias)
{
    int idx = blockIdx.x * blockDim.x + threadIdx.x;
    if (idx >= N_NODES * C) return;
    const int n = idx / C;
    const int c = idx - n * C;
    const float di = dinv[n];
    out[idx] = hw[idx] * di * di + bias[c];
}

template <int C>
__global__ void agg_edges_kernel(float* __restrict__ out, const float* __restrict__ hw,
                                 const int* __restrict__ src, const int* __restrict__ dst,
                                 const float* __restrict__ norm)
{
    constexpr int QC = C / 4;                       // float4 chunks per edge
    int idx = blockIdx.x * blockDim.x + threadIdx.x;
    if (idx >= N_EDGES * QC) return;
    const int e = idx / QC;
    const int q = idx - e * QC;
    const int s = src[e];
    const int d = dst[e];
    const float nrm = norm[e];
    const float4 v = *(const float4*)(hw + (size_t)s * C + q * 4);
    float* o = out + (size_t)d * C + q * 4;
    atomicAdd(o + 0, v.x * nrm);
    atomicAdd(o + 1, v.y * nrm);
    atomicAdd(o + 2, v.z * nrm);
    atomicAdd(o + 3, v.w * nrm);
}

__global__ void leaky_relu_kernel(float* __restrict__ h, int n)
{
    int i = blockIdx.x * blockDim.x + threadIdx.x;
    if (i < n) {
        const float x = h[i];
        h[i] = x > 0.0f ? x : 0.2f * x;
    }
}

// ---------------------------------------------------------------------------
// Host launcher
// ---------------------------------------------------------------------------
extern "C" void kernel_launch(void* const* d_in, const int* in_sizes, int n_in,
                              void* d_out, int out_size, void* d_ws, size_t ws_size,
                              hipStream_t stream)
{
    const float* x    = (const float*)d_in[0];                 // [N, IN_C]
    const int*   ei   = (const int*)d_in[1];                   // [2, E]
    const float* W_in = (const float*)d_in[2];                 // [IN_C, HID]
    const float* b_in = (const float*)d_in[3];                 // [HID]
    const float* Wg   = (const float*)d_in[4];                 // [L, HID, HID]
    const float* bg   = (const float*)d_in[5];                 // [L, HID]
    const float* Ws   = (const float*)d_in[6];                 // [L, HID, LAT]
    const float* bs   = (const float*)d_in[7];                 // [L, LAT]
    const float* Wl   = (const float*)d_in[8];                 // [HID, LAT]
    const float* bl   = (const float*)d_in[9];                 // [LAT]

    const int* src = ei;
    const int* dst = ei + N_EDGES;

    // Workspace layout (all fully overwritten every call -> graph-capture safe)
    float* ws   = (float*)d_ws;
    float* dinv = ws;                                  // N
    float* norm = dinv + N_NODES;                      // E
    float* h0   = norm + N_EDGES;                      // N*HID
    float* h1   = h0 + (size_t)N_NODES * HID;
    float* h2   = h1 + (size_t)N_NODES * HID;
    float* h3   = h2 + (size_t)N_NODES * HID;
    float* tmp  = h3 + (size_t)N_NODES * HID;          // N*HID scratch (GEMM out)
    float* hbuf[4] = {h0, h1, h2, h3};
    float* out = (float*)d_out;                        // [N, LAT]

    const int TB = 256;
    const int gN  = (N_NODES + TB - 1) / TB;
    const int gE  = (N_EDGES + TB - 1) / TB;
    const int gNH = (N_NODES * HID + TB - 1) / TB;
    const int gNL = (N_NODES * LAT + TB - 1) / TB;
    const int gEH = (N_EDGES * (HID / 4) + TB - 1) / TB;
    const int gEL = (N_EDGES * (LAT / 4) + TB - 1) / TB;

    // --- GCN symmetric degree normalization ---
    deg_init_kernel<<<gN, TB, 0, stream>>>(dinv);
    deg_count_kernel<<<gE, TB, 0, stream>>>(dinv, dst);
    deg_rsqrt_kernel<<<gN, TB, 0, stream>>>(dinv);
    edge_norm_kernel<<<gE, TB, 0, stream>>>(norm, dinv, src, dst);

    // --- input projection: h0 = x @ W_in + b_in ---
    wmma_gemm_kernel<IN_C, HID, false>
        <<<dim3(GEMM_GX, HID / 16), TB, 0, stream>>>(x, W_in, b_in, h0);

    // --- 3 GCN layers: h_{i+1} = leaky_relu(agg(h_i @ Wg_i) + bg_i) ---
    for (int i = 0; i < L; ++i) {
        wmma_gemm_kernel<HID, HID, false>
            <<<dim3(GEMM_GX, HID / 16), TB, 0, stream>>>(hbuf[i], Wg + (size_t)i * HID * HID,
                                                         nullptr, tmp);
        agg_init_kernel<HID><<<gNH, TB, 0, stream>>>(hbuf[i + 1], tmp, dinv, bg + (size_t)i * HID);
        agg_edges_kernel<HID><<<gEH, TB, 0, stream>>>(hbuf[i + 1], tmp, src, dst, norm);
        leaky_relu_kernel<<<gNH, TB, 0, stream>>>(hbuf[i + 1], N_NODES * HID);
    }

    // --- latent = agg(h3 @ Wl) + bl ---
    wmma_gemm_kernel<HID, LAT, false>
        <<<dim3(GEMM_GX, LAT / 16), TB, 0, stream>>>(h3, Wl, nullptr, tmp);
    agg_init_kernel<LAT><<<gNL, TB, 0, stream>>>(out, tmp, dinv, bl);
    agg_edges_kernel<LAT><<<gEL, TB, 0, stream>>>(out, tmp, src, dst, norm);

    // --- skip connections: latent += h_i @ Ws_i + bs_i ---
    for (int i = 0; i < L; ++i) {
        wmma_gemm_kernel<HID, LAT, true>
            <<<dim3(GEMM_GX, LAT / 16), TB, 0, stream>>>(hbuf[i], Ws + (size_t)i * HID * LAT,
                                                         bs + (size_t)i * LAT, out);
    }
}